// Sampler_66486093742538
// MI455X (gfx1250) — compile-verified
//
#include <hip/hip_runtime.h>
#include <hip/hip_bf16.h>

#ifndef __has_builtin
#define __has_builtin(x) 0
#endif

#define TEMP_INV  (1.0f / 0.7f)
#define TOP_P_F   0.9f
#define TOP_K_N   50u
#define B_ROWS    256
#define VOCAB     128000
#define NTHREADS  256
#define NCHUNK    (VOCAB / (NTHREADS * 4))   // 125 float4 chunks per thread

#if defined(__AMDGCN__) && __has_builtin(__builtin_amdgcn_global_load_async_to_lds_b128)
#define HAVE_ASYNC 1
#else
#define HAVE_ASYNC 0
#endif

#if defined(__AMDGCN__) && __has_builtin(__builtin_amdgcn_global_store_async_from_lds_b128)
#define HAVE_ASYNC_ST 1
#else
#define HAVE_ASYNC_ST 0
#endif

#if HAVE_ASYNC || HAVE_ASYNC_ST
typedef int v4i __attribute__((ext_vector_type(4)));
typedef __attribute__((address_space(1))) v4i* gptr_v4i;
typedef __attribute__((address_space(3))) v4i* lptr_v4i;

#if __has_builtin(__builtin_amdgcn_s_wait_asynccnt)
#define WAIT_ASYNC(n) do { __builtin_amdgcn_s_wait_asynccnt(n); \
                           asm volatile("" ::: "memory"); } while (0)
#else
#define WAIT_ASYNC(n) asm volatile("s_wait_asynccnt %0" :: "i"(n) : "memory")
#endif

#if __has_builtin(__builtin_amdgcn_s_wait_dscnt)
#define WAIT_DS0() do { __builtin_amdgcn_s_wait_dscnt(0); \
                        asm volatile("" ::: "memory"); } while (0)
#else
#define WAIT_DS0() asm volatile("s_wait_dscnt 0" ::: "memory")
#endif
#endif

// Order-preserving float -> u32 key (monotonic: -inf < ... < +inf)
__device__ __forceinline__ unsigned fkey(float f) {
    unsigned u = __float_as_uint(f);
    return (u & 0x80000000u) ? ~u : (u | 0x80000000u);
}

__global__ __launch_bounds__(NTHREADS)
void sampler_row_kernel(const float* __restrict__ logits,
                        const float* __restrict__ noise,
                        float* __restrict__ out)
{
    const int row = blockIdx.x;
    const int tid = threadIdx.x;
    const float4* __restrict__ lg4 = (const float4*)(logits + (size_t)row * VOCAB);
    const float4* __restrict__ un4 = (const float4*)(noise  + (size_t)row * VOCAB);
    float* __restrict__ probs = out + B_ROWS + (size_t)row * VOCAB;

    __shared__ float    s_m[NTHREADS];
    __shared__ float    s_s[NTHREADS];
    __shared__ int      s_i[NTHREADS];
    __shared__ unsigned s_cnt[256];
    __shared__ float    s_w[256];
#if HAVE_ASYNC || HAVE_ASYNC_ST
    __shared__ float    s_stage[2][NTHREADS * 4];   // 8 KB double buffer
#endif

    // ---------------- Phase A: streaming online max + sum(exp(x - max)) ----
    // Async LDS-DMA staged, double-buffered: global -> LDS (ASYNCcnt) -> VGPR.
    float m = -INFINITY, s = 0.0f;
#if HAVE_ASYNC
    {
        __builtin_amdgcn_global_load_async_to_lds_b128(
            (gptr_v4i)&lg4[tid], (lptr_v4i)&s_stage[0][tid * 4], 0, 0);
        for (int c = 0; c < NCHUNK; ++c) {
            if (c + 1 < NCHUNK) {
                WAIT_DS0();   // prior ds_reads of the target buffer retired
                __builtin_amdgcn_global_load_async_to_lds_b128(
                    (gptr_v4i)&lg4[(c + 1) * NTHREADS + tid],
                    (lptr_v4i)&s_stage[(c + 1) & 1][tid * 4], 0, 0);
                WAIT_ASYNC(1);
            } else {
                WAIT_ASYNC(0);
            }
            const float4 v = *(const float4*)&s_stage[c & 1][tid * 4];
            float xs[4] = {v.x, v.y, v.z, v.w};
            #pragma unroll
            for (int j = 0; j < 4; ++j) {
                float x = xs[j] * TEMP_INV;
                if (x > m) { s = s * __expf(m - x) + 1.0f; m = x; }
                else       { s += __expf(x - m); }
            }
        }
    }
#else
    for (int c = 0; c < NCHUNK; ++c) {
        const float4 v = lg4[c * NTHREADS + tid];
        float xs[4] = {v.x, v.y, v.z, v.w};
        #pragma unroll
        for (int j = 0; j < 4; ++j) {
            float x = xs[j] * TEMP_INV;
            if (x > m) { s = s * __expf(m - x) + 1.0f; m = x; }
            else       { s += __expf(x - m); }
        }
    }
#endif
    s_m[tid] = m; s_s[tid] = s;
    __syncthreads();
    for (int off = NTHREADS / 2; off > 0; off >>= 1) {
        if (tid < off) {
            float ma = s_m[tid], mb = s_m[tid + off];
            float m2 = fmaxf(ma, mb);
            s_s[tid] = s_s[tid] * __expf(ma - m2) + s_s[tid + off] * __expf(mb - m2);
            s_m[tid] = m2;
        }
        __syncthreads();
    }
    const float M = s_m[0];
    const float Z = s_s[0];
    __syncthreads();

    // ---------------- Phase B: fused radix descent for both thresholds -----
    // top-k: 50th-largest key (count histogram, LDS ds_add atomics)
    // top-p: boundary key k* s.t. weight(keys > k*) < 0.9*Z <= weight(keys >= k*)
    const float W9 = TOP_P_F * Z;
    unsigned kpref = 0, ppref = 0;
    unsigned kr = TOP_K_N;
    float pA = 0.0f;   // exp-weight strictly above current top-p prefix range

    for (int L = 0; L < 4; ++L) {
        const int shift = 24 - 8 * L;
        s_cnt[tid] = 0u;
        s_w[tid]   = 0.0f;
        __syncthreads();
        for (int c = 0; c < NCHUNK; ++c) {
            if (c + 1 < NCHUNK)
                __builtin_prefetch(&lg4[(c + 1) * NTHREADS + tid], 0, 1);
            const float4 v = lg4[c * NTHREADS + tid];
            float xs[4] = {v.x, v.y, v.z, v.w};
            #pragma unroll
            for (int j = 0; j < 4; ++j) {
                float x = xs[j] * TEMP_INV;
                unsigned key = fkey(x);
                unsigned bin = (key >> shift) & 0xFFu;
                bool okk = (L == 0) || ((key >> (shift + 8)) == kpref);
                bool okp = (L == 0) || ((key >> (shift + 8)) == ppref);
                if (okk) atomicAdd(&s_cnt[bin], 1u);
                if (okp) atomicAdd(&s_w[bin], __expf(x - M));
            }
        }
        __syncthreads();
        // Redundant identical serial scans (LDS-broadcast reads); keep rolled.
        {
            unsigned cum = 0; int kb = 0;
            #pragma unroll 1
            for (int b = 255; b >= 0; --b) {
                unsigned c = s_cnt[b];
                if (cum + c >= kr) { kb = b; break; }
                cum += c;
            }
            kr -= cum;
            kpref = (kpref << 8) | (unsigned)kb;

            float a = pA; int pb = 0;
            #pragma unroll 1
            for (int b = 255; b >= 0; --b) {
                float w = s_w[b];
                if (a + w >= W9) { pb = b; break; }
                a += w;
            }
            pA = a;
            ppref = (ppref << 8) | (unsigned)pb;
        }
        __syncthreads();
    }
    // keep iff key >= T  (covers both n_keep<k and n_keep>=k cases exactly)
    const unsigned T = (kpref > ppref) ? kpref : ppref;

    // ---------------- Phase C: kept-sum Z' and Gumbel argmax ---------------
    float zk = 0.0f, best = -INFINITY;
    int bidx = 0;
    for (int c = 0; c < NCHUNK; ++c) {
        const int f4i = c * NTHREADS + tid;
        const float4 v = lg4[f4i];
        const float4 u = un4[f4i];
        float xs[4] = {v.x, v.y, v.z, v.w};
        float us[4] = {u.x, u.y, u.z, u.w};
        #pragma unroll
        for (int j = 0; j < 4; ++j) {
            float x = xs[j] * TEMP_INV;
            if (fkey(x) >= T) {
                zk += __expf(x - M);
                float g  = -__logf(-__logf(us[j]));   // Gumbel(0,1)
                float sc = x + g;
                if (sc > best) { best = sc; bidx = f4i * 4 + j; }
            }
        }
    }
    s_s[tid] = zk; s_m[tid] = best; s_i[tid] = bidx;
    __syncthreads();
    for (int off = NTHREADS / 2; off > 0; off >>= 1) {
        if (tid < off) {
            s_s[tid] += s_s[tid + off];
            float vb = s_m[tid + off];
            if (vb > s_m[tid] || (vb == s_m[tid] && s_i[tid + off] < s_i[tid])) {
                s_m[tid] = vb; s_i[tid] = s_i[tid + off];
            }
        }
        __syncthreads();
    }
    const float invZk = 1.0f / s_s[0];
    if (tid == 0) out[row] = (float)s_i[0];
    __syncthreads();

    // ---------------- Phase D: write probs ---------------------------------
    // Async store path: VGPR -> LDS (ds_store_b128) -> global (ASYNCcnt DMA),
    // double-buffered; falls back to direct b128 stores if builtin absent.
#if HAVE_ASYNC_ST
    for (int c = 0; c < NCHUNK; ++c) {
        const int f4i = c * NTHREADS + tid;
        const float4 v = lg4[f4i];
        float xs[4] = {v.x, v.y, v.z, v.w};
        float ps[4];
        #pragma unroll
        for (int j = 0; j < 4; ++j) {
            float x = xs[j] * TEMP_INV;
            ps[j] = (fkey(x) >= T) ? __expf(x - M) * invZk : 0.0f;
        }
        // buffer reuse guard: store issued 2 iterations ago must be complete
        WAIT_ASYNC(1);
        float4* slot = (float4*)&s_stage[c & 1][tid * 4];
        float4 o; o.x = ps[0]; o.y = ps[1]; o.z = ps[2]; o.w = ps[3];
        *slot = o;
        WAIT_DS0();   // ds_store visible to the async-DMA engine
        __builtin_amdgcn_global_store_async_from_lds_b128(
            (gptr_v4i)&((float4*)probs)[f4i], (lptr_v4i)slot, 0, 0);
    }
    WAIT_ASYNC(0);
#else
    float4* __restrict__ pr4 = (float4*)probs;
    for (int c = 0; c < NCHUNK; ++c) {
        const int f4i = c * NTHREADS + tid;
        const float4 v = lg4[f4i];
        float xs[4] = {v.x, v.y, v.z, v.w};
        float ps[4];
        #pragma unroll
        for (int j = 0; j < 4; ++j) {
            float x = xs[j] * TEMP_INV;
            ps[j] = (fkey(x) >= T) ? __expf(x - M) * invZk : 0.0f;
        }
        float4 o; o.x = ps[0]; o.y = ps[1]; o.z = ps[2]; o.w = ps[3];
        pr4[f4i] = o;
    }
#endif
}

extern "C" void kernel_launch(void* const* d_in, const int* in_sizes, int n_in,
                              void* d_out, int out_size, void* d_ws, size_t ws_size,
                              hipStream_t stream) {
    (void)in_sizes; (void)n_in; (void)d_ws; (void)ws_size; (void)out_size;
    // setup_inputs order: input_ids (unused), logits [B,V] f32, uniform_noise [B,V] f32
    const float* logits = (const float*)d_in[1];
    const float* noise  = (const float*)d_in[2];
    float* out = (float*)d_out;   // [B] token-as-float  ++  [B,V] probs
    hipLaunchKernelGGL(sampler_row_kernel, dim3(B_ROWS), dim3(NTHREADS), 0, stream,
                       logits, noise, out);
}